// GATGraphLayer_51943334478494
// MI455X (gfx1250) — compile-verified
//
#include <hip/hip_runtime.h>
#include <hip/hip_bf16.h>
#include <math.h>

#define NNODES  400000
#define MFEAT   256
#define NHEADS  4
#define NGRAPH  8192
#define CHUNK   64         // rows staged in LDS per iteration (avg segment ~49 rows -> 1 chunk)
#define TSTRIDE 260        // padded LDS row stride (floats): 260 % 64 == 4 -> conflict-free
#define WSTRIDE 260
#define BLOCK   256        // 8 wave32s

typedef __attribute__((ext_vector_type(2))) float v2f;
typedef __attribute__((ext_vector_type(8))) float v8f;

#if __has_builtin(__builtin_amdgcn_global_load_async_to_lds_b128) && \
    __has_builtin(__builtin_amdgcn_s_wait_asynccnt)
#define USE_ASYNC_LDS 1
// Builtin prototype (from hipcc diagnostic): (v4i AS1*, v4i AS3*, int, int)
typedef int v4i __attribute__((vector_size(16)));
typedef __attribute__((address_space(1))) v4i as1_v4i;
typedef __attribute__((address_space(3))) v4i as3_v4i;
#else
#define USE_ASYNC_LDS 0
#endif

__device__ __forceinline__ int lower_bound_i32(const int* a, int n, int key) {
  int lo = 0, hi = n;
  while (lo < hi) { int mid = (lo + hi) >> 1; if (a[mid] < key) lo = mid + 1; else hi = mid; }
  return lo;
}

// wave32 xor-lane float add via ds_swizzle (group-of-32: offset[14:10]=xor, [4:0]=and)
__device__ __forceinline__ float swz_add_x1(float s) {
  return s + __builtin_bit_cast(float,
      __builtin_amdgcn_ds_swizzle(__builtin_bit_cast(int, s), 0x041F));
}
__device__ __forceinline__ float swz_add_x2(float s) {
  return s + __builtin_bit_cast(float,
      __builtin_amdgcn_ds_swizzle(__builtin_bit_cast(int, s), 0x081F));
}

__global__ __launch_bounds__(BLOCK)
void gat_gate_segment_kernel(const float* __restrict__ feats,
                             const float* __restrict__ W,
                             const float* __restrict__ bias,
                             const int*   __restrict__ seg_ids,
                             float*       __restrict__ out) {
  extern __shared__ float smem[];
  float* tile  = smem;                     // CHUNK * TSTRIDE floats (~65KB -> 4 WGs / WGP)
  float* gmean = smem + CHUNK * TSTRIDE;   // CHUNK floats (mean-over-heads gate per row)
  __shared__ float ldsW[5 * WSTRIDE];      // 4 head rows + 1 zero pad row
  __shared__ int   range[2];

  const int t    = threadIdx.x;
  const int lane = t & 31;
  const int wave = t >> 5;
  const int seg  = blockIdx.x;

  // Stage W into LDS; row 4 is an all-zero pad row read by lanes with head >= 4.
  for (int i = t; i < 5 * WSTRIDE; i += BLOCK) {
    const int r = i / WSTRIDE, c = i - r * WSTRIDE;
    ldsW[i] = (r < NHEADS && c < MFEAT) ? W[r * MFEAT + c] : 0.0f;
  }
  if (t == 0) range[0] = lower_bound_i32(seg_ids, NNODES, seg);
  if (t == 1) range[1] = lower_bound_i32(seg_ids, NNODES, seg + 1);
  __syncthreads();

  const int row_start = range[0];
  const int row_end   = range[1];

  // WMMA per-lane operand roles (ISA 16x4 A / 4x16 B f32 layouts):
  //   A: lane holds M = lane&15, K sub-offset = (lane>>4)*2 (+0,+1)
  //   B: lane holds N = lane&15 (head), same K sub-offset
  const int   mrow  = lane & 15;
  const int   koff  = (lane >> 4) << 1;
  const int   ncol  = lane & 15;
  const float bn    = (ncol < NHEADS) ? bias[ncol] : 0.0f;
  const float hmask = (ncol < NHEADS) ? (1.0f / NHEADS) : 0.0f;  // fold head-mean scale
  const float* brow = &ldsW[(ncol < NHEADS ? ncol : NHEADS) * WSTRIDE];

  float accs = 0.0f;
  float accm = -INFINITY;

  for (int cs = row_start; cs < row_end; cs += CHUNK) {
    const int rows = min(CHUNK, row_end - cs);

    // ---- stage chunk into LDS: read feats exactly once from HBM
    const float4* src = (const float4*)(feats + (size_t)cs * MFEAT);
    const int n4 = rows * (MFEAT / 4);
#if USE_ASYNC_LDS
    // CDNA5 async copy: global -> LDS without VGPR round-trip, tracked by ASYNCcnt
    for (int i = t; i < n4; i += BLOCK) {
      const int r = i >> 6, c4 = i & 63;
      __builtin_amdgcn_global_load_async_to_lds_b128(
          (as1_v4i*)(src + i),
          (as3_v4i*)&tile[r * TSTRIDE + (c4 << 2)], 0, 0);
    }
#else
    for (int i = t; i < n4; i += BLOCK) {
      const int r = i >> 6, c4 = i & 63;
      float4 v = src[i];
      *(float4*)&tile[r * TSTRIDE + (c4 << 2)] = v;
    }
#endif
    // Pull the next chunk into L2 while we compute on this one (global_prefetch_b8).
    if (cs + CHUNK < row_end) {
      const char* nxt = (const char*)(feats + (size_t)(cs + CHUNK) * MFEAT);
      __builtin_prefetch(nxt + t * 256, 0, 0);
    }
#if USE_ASYNC_LDS
    __builtin_amdgcn_s_wait_asynccnt(0);
#endif
    __syncthreads();

    // ---- phase A: each wave does one 16-row gate tile via V_WMMA_F32_16X16X4_F32,
    // accumulating K=256 in 64 steps. Wave-uniform guard keeps EXEC all-ones for WMMA.
    const int tile0 = wave * 16;
    if (tile0 < rows) {
      const float* arow = &tile[(tile0 + mrow) * TSTRIDE];
      v8f acc = {0.f, 0.f, 0.f, 0.f, 0.f, 0.f, 0.f, 0.f};
      #pragma unroll 8
      for (int kk = 0; kk < MFEAT / 4; ++kk) {
        v2f a  = *(const v2f*)&arow[(kk << 2) + koff];
        v2f bv = *(const v2f*)&brow[(kk << 2) + koff];
        acc = __builtin_amdgcn_wmma_f32_16x16x4_f32(false, a, false, bv,
                                                    (short)0, acc, false, false);
      }
      // gates: sigmoid(dot + b[head]); branchless mask, then xor-lane head reduce.
      // D layout: VGPR i -> (M=i, N=lane) lanes 0-15, (M=8+i, N=lane-16) lanes 16-31.
      #pragma unroll
      for (int i = 0; i < 8; ++i) {
        const float e = __expf(-(acc[i] + bn));
        float s = hmask * __builtin_amdgcn_rcpf(1.0f + e);  // v_exp + v_rcp only
        s = swz_add_x1(s);
        s = swz_add_x2(s);   // lanes 0 / 16 now hold 0.25 * sum over the 4 heads
        if ((lane & 15) == 0)
          gmean[tile0 + ((lane >> 4) << 3) + i] = s;
      }
    }
    __syncthreads();

    // ---- phase B: fused gated segment-sum + segment-max, one feature column / thread
    #pragma unroll 4
    for (int r = 0; r < rows; ++r) {
      const float x = tile[r * TSTRIDE + t];
      accs = fmaf(x, gmean[r], accs);
      accm = fmaxf(accm, x);
    }
    __syncthreads();
  }

  // out = concat([segment_sum(feats * mean_gate), segment_max(feats)], axis=1)
  out[(size_t)seg * (2 * MFEAT) + t]         = accs;
  out[(size_t)seg * (2 * MFEAT) + MFEAT + t] = accm;
}

extern "C" void kernel_launch(void* const* d_in, const int* in_sizes, int n_in,
                              void* d_out, int out_size, void* d_ws, size_t ws_size,
                              hipStream_t stream) {
  (void)in_sizes; (void)n_in; (void)d_ws; (void)ws_size; (void)out_size;
  const float* feats   = (const float*)d_in[0];
  const float* W       = (const float*)d_in[1];
  const float* bias    = (const float*)d_in[2];
  const int*   seg_ids = (const int*)d_in[3];
  float*       out     = (float*)d_out;

  const size_t smem_bytes = (size_t)(CHUNK * TSTRIDE + CHUNK) * sizeof(float);
  gat_gate_segment_kernel<<<NGRAPH, BLOCK, smem_bytes, stream>>>(feats, W, bias, seg_ids, out);
}